// FullCovLDAHead_54116587929771
// MI455X (gfx1250) — compile-verified
//
#include <hip/hip_runtime.h>
#include <math.h>

#define D 512
#define BB 128
#define CC 1000

typedef __attribute__((ext_vector_type(2))) float v2f;
typedef __attribute__((ext_vector_type(8))) float v8f;

__device__ __forceinline__ float softplus_f(float x) {
  return (x > 20.f) ? x : log1pf(expf(x));
}

// ---------------------------------------------------------------------------
// Kernel 1: block 0 -> safe diag + logdet ; block 1 -> logsumexp(prior)
// ---------------------------------------------------------------------------
__global__ void prep_kernel(const float* __restrict__ raw,
                            const float* __restrict__ prior,
                            float* __restrict__ diag,
                            float* __restrict__ scal) {
  __shared__ float red[256];
  const int t = threadIdx.x;
  if (blockIdx.x == 0) {
    float s = 0.f;
    for (int d = t; d < D; d += 256) {
      float v = softplus_f(raw[(size_t)d * D + d]) + 1e-4f;
      diag[d] = v;
      s += logf(v);
    }
    red[t] = s; __syncthreads();
    for (int o = 128; o > 0; o >>= 1) { if (t < o) red[t] += red[t + o]; __syncthreads(); }
    if (t == 0) scal[0] = 2.f * red[0];
  } else {
    float m = -1e30f;
    for (int c = t; c < CC; c += 256) m = fmaxf(m, prior[c]);
    red[t] = m; __syncthreads();
    for (int o = 128; o > 0; o >>= 1) { if (t < o) red[t] = fmaxf(red[t], red[t + o]); __syncthreads(); }
    m = red[0]; __syncthreads();
    float s = 0.f;
    for (int c = t; c < CC; c += 256) s += expf(prior[c] - m);
    red[t] = s; __syncthreads();
    for (int o = 128; o > 0; o >>= 1) { if (t < o) red[t] += red[t + o]; __syncthreads(); }
    if (t == 0) scal[1] = m + logf(red[0]);
  }
}

// ---------------------------------------------------------------------------
// Kernel 2: invert lower-triangular L, one wave per column j.
// Writes A^T row-major: AinvT[j*D + i] = (L^-1)[i, j]   (coalesced row write)
// ---------------------------------------------------------------------------
__global__ void trinv_kernel(const float* __restrict__ raw,
                             const float* __restrict__ diag,
                             float* __restrict__ AinvT) {
  __shared__ float xs[D];
  const int j = blockIdx.x;
  const int lane = threadIdx.x;      // 32 threads = 1 wave
  if (lane == 0) xs[j] = 1.f / diag[j];
  __syncthreads();
  for (int i = j + 1; i < D; ++i) {
    const float* Lrow = raw + (size_t)i * D;   // strictly-lower part used only
    float s = 0.f;
    for (int k = j + lane; k < i; k += 32) s += Lrow[k] * xs[k];
    for (int o = 16; o > 0; o >>= 1) s += __shfl_xor(s, o, 32);
    if (lane == 0) xs[i] = -s / diag[i];
    __syncthreads();
  }
  float* row = AinvT + (size_t)j * D;
  for (int i = lane; i < D; i += 32) row[i] = (i < j) ? 0.f : xs[i];
}

// ---------------------------------------------------------------------------
// Kernel 3: SZ = z @ AinvT   (M=128, N=512, K=512), f32 WMMA 16x16x4.
// One wave computes a 16x64 strip (4 accumulators), reusing the A fragment
// across 4 N-tiles. 8 m-tiles x 8 n-groups = 64 waves -> 8 blocks x 8 waves.
// ---------------------------------------------------------------------------
__global__ void gemm_sz_kernel(const float* __restrict__ Z,
                               const float* __restrict__ BT,
                               float* __restrict__ SZ) {
  const int wave = (blockIdx.x * blockDim.x + threadIdx.x) >> 5;
  const int lane = threadIdx.x & 31;
  const int g = lane >> 4, l = lane & 15;
  const int m0 = (wave >> 3) * 16;
  const int n0 = (wave & 7) * 64;
  v8f acc0 = {}, acc1 = {}, acc2 = {}, acc3 = {};
  for (int k0 = 0; k0 < D; k0 += 4) {
    v2f a;
    const float* ar = Z + (size_t)(m0 + l) * D + k0 + 2 * g;
    a.x = ar[0];
    a.y = ar[1];
    const float* b0 = BT + (size_t)(k0 + 2 * g) * D + n0 + l;
    const float* b1 = b0 + D;
    v2f b;
#define SZ_STEP(ACC, Q) \
    b.x = b0[16 * Q]; b.y = b1[16 * Q]; \
    ACC = __builtin_amdgcn_wmma_f32_16x16x4_f32(false, a, false, b, (short)0, ACC, false, false);
    SZ_STEP(acc0, 0) SZ_STEP(acc1, 1) SZ_STEP(acc2, 2) SZ_STEP(acc3, 3)
#undef SZ_STEP
  }
  for (int r = 0; r < 8; ++r) {
    float* orow = SZ + (size_t)(m0 + r + 8 * g) * D + n0 + l;
    orow[0]  = acc0[r];
    orow[16] = acc1[r];
    orow[32] = acc2[r];
    orow[48] = acc3[r];
  }
}

// ---------------------------------------------------------------------------
// Kernel 4: SM = mu @ AinvT  (M=1000 padded to 1008, N=512, K=512).
// 16x64 per wave: 63 m-tiles x 8 n-groups = 504 waves -> 63 blocks x 8 waves.
// ---------------------------------------------------------------------------
__global__ void gemm_sm_kernel(const float* __restrict__ MU,
                               const float* __restrict__ BT,
                               float* __restrict__ SM) {
  const int wave = (blockIdx.x * blockDim.x + threadIdx.x) >> 5;
  const int lane = threadIdx.x & 31;
  const int g = lane >> 4, l = lane & 15;
  const int m0 = (wave >> 3) * 16;
  const int n0 = (wave & 7) * 64;
  const int m = m0 + l;
  const bool mv = (m < CC);
  v8f acc0 = {}, acc1 = {}, acc2 = {}, acc3 = {};
  for (int k0 = 0; k0 < D; k0 += 4) {
    v2f a;
    const float* ar = MU + (size_t)m * D + k0 + 2 * g;
    a.x = mv ? ar[0] : 0.f;
    a.y = mv ? ar[1] : 0.f;
    const float* b0 = BT + (size_t)(k0 + 2 * g) * D + n0 + l;
    const float* b1 = b0 + D;
    v2f b;
#define SM_STEP(ACC, Q) \
    b.x = b0[16 * Q]; b.y = b1[16 * Q]; \
    ACC = __builtin_amdgcn_wmma_f32_16x16x4_f32(false, a, false, b, (short)0, ACC, false, false);
    SM_STEP(acc0, 0) SM_STEP(acc1, 1) SM_STEP(acc2, 2) SM_STEP(acc3, 3)
#undef SM_STEP
  }
  for (int r = 0; r < 8; ++r) {
    const int mr = m0 + r + 8 * g;
    if (mr < CC) {
      float* orow = SM + (size_t)mr * D + n0 + l;
      orow[0]  = acc0[r];
      orow[16] = acc1[r];
      orow[32] = acc2[r];
      orow[48] = acc3[r];
    }
  }
}

// ---------------------------------------------------------------------------
// Kernel 5: squared row norms of SZ (128 rows) and SM (1000 rows).
// 1128 waves -> 141 blocks x 8 waves.
// ---------------------------------------------------------------------------
__global__ void norms_kernel(const float* __restrict__ SZ,
                             const float* __restrict__ SM,
                             float* __restrict__ nz,
                             float* __restrict__ nm) {
  const int wave = (blockIdx.x * blockDim.x + threadIdx.x) >> 5;
  const int lane = threadIdx.x & 31;
  const float* row;
  float* dst;
  if (wave < BB)            { row = SZ + (size_t)wave * D;        dst = nz + wave; }
  else if (wave < BB + CC)  { row = SM + (size_t)(wave - BB) * D; dst = nm + (wave - BB); }
  else return;
  float s = 0.f;
  for (int k = lane; k < D; k += 32) { float v = row[k]; s += v * v; }
  for (int o = 16; o > 0; o >>= 1) s += __shfl_xor(s, o, 32);
  if (lane == 0) *dst = s;
}

// ---------------------------------------------------------------------------
// Kernel 6: G = SZ @ SM^T (M=128, N=1000 padded to 1024, K=512) with fused
// epilogue: out[b,c] = (prior[c]-lse) - 0.5*(nz[b] + nm[c] - 2*G + logdet).
// 16x64 per wave: 8 m-tiles x 16 n-groups = 128 waves -> 16 blocks x 8 waves.
// ---------------------------------------------------------------------------
__global__ void gemm_out_kernel(const float* __restrict__ SZ,
                                const float* __restrict__ SM,
                                const float* __restrict__ nz,
                                const float* __restrict__ nm,
                                const float* __restrict__ prior,
                                const float* __restrict__ scal,
                                float* __restrict__ out) {
  const int wave = (blockIdx.x * blockDim.x + threadIdx.x) >> 5;
  const int lane = threadIdx.x & 31;
  const int g = lane >> 4, l = lane & 15;
  const int m0 = (wave >> 4) * 16;
  const int n0 = (wave & 15) * 64;
  const int n1 = n0 + l, n2 = n1 + 16, n3 = n1 + 32, n4 = n1 + 48;
  const bool v1 = (n1 < CC), v2 = (n2 < CC), v3 = (n3 < CC), v4 = (n4 < CC);
  v8f acc0 = {}, acc1 = {}, acc2 = {}, acc3 = {};
  for (int k0 = 0; k0 < D; k0 += 4) {
    v2f a;
    const float* ar = SZ + (size_t)(m0 + l) * D + k0 + 2 * g;
    a.x = ar[0];
    a.y = ar[1];
    const size_t ko = (size_t)(k0 + 2 * g);
    v2f b;
#define GO_STEP(ACC, NQ, VQ) \
    { const float* br = SM + (size_t)(NQ) * D + ko; \
      b.x = (VQ) ? br[0] : 0.f; \
      b.y = (VQ) ? br[1] : 0.f; \
      ACC = __builtin_amdgcn_wmma_f32_16x16x4_f32(false, a, false, b, (short)0, ACC, false, false); }
    GO_STEP(acc0, n1, v1) GO_STEP(acc1, n2, v2) GO_STEP(acc2, n3, v3) GO_STEP(acc3, n4, v4)
#undef GO_STEP
  }
  const float logdet = scal[0];
  const float lse = scal[1];
  const float lp1 = v1 ? (prior[n1] - lse) : 0.f;
  const float lp2 = v2 ? (prior[n2] - lse) : 0.f;
  const float lp3 = v3 ? (prior[n3] - lse) : 0.f;
  const float lp4 = v4 ? (prior[n4] - lse) : 0.f;
  const float nm1 = v1 ? nm[n1] : 0.f;
  const float nm2 = v2 ? nm[n2] : 0.f;
  const float nm3 = v3 ? nm[n3] : 0.f;
  const float nm4 = v4 ? nm[n4] : 0.f;
  for (int r = 0; r < 8; ++r) {
    const int m = m0 + r + 8 * g;
    const float nzm = nz[m];
    float* orow = out + (size_t)m * CC;
    if (v1) orow[n1] = lp1 - 0.5f * (nzm + nm1 - 2.f * acc0[r] + logdet);
    if (v2) orow[n2] = lp2 - 0.5f * (nzm + nm2 - 2.f * acc1[r] + logdet);
    if (v3) orow[n3] = lp3 - 0.5f * (nzm + nm3 - 2.f * acc2[r] + logdet);
    if (v4) orow[n4] = lp4 - 0.5f * (nzm + nm4 - 2.f * acc3[r] + logdet);
  }
}

// ---------------------------------------------------------------------------
extern "C" void kernel_launch(void* const* d_in, const int* in_sizes, int n_in,
                              void* d_out, int out_size, void* d_ws, size_t ws_size,
                              hipStream_t stream) {
  (void)in_sizes; (void)n_in; (void)out_size; (void)ws_size;
  const float* z     = (const float*)d_in[0];   // (128, 512)
  const float* mu    = (const float*)d_in[1];   // (1000, 512)
  const float* raw   = (const float*)d_in[2];   // (512, 512)
  const float* prior = (const float*)d_in[3];   // (1000,)
  float* out = (float*)d_out;                   // (128, 1000)

  float* ws    = (float*)d_ws;                  // ~3.4 MB of f32 scratch
  float* diag  = ws;                            // 512
  float* scal  = ws + 512;                      // [0]=logdet, [1]=lse
  float* AinvT = ws + 1024;                     // 512*512
  float* SZ    = AinvT + (size_t)D * D;         // 128*512
  float* SM    = SZ + (size_t)BB * D;           // 1000*512
  float* nz    = SM + (size_t)CC * D;           // 128
  float* nm    = nz + BB;                       // 1000

  prep_kernel<<<2, 256, 0, stream>>>(raw, prior, diag, scal);
  trinv_kernel<<<D, 32, 0, stream>>>(raw, diag, AinvT);
  gemm_sz_kernel<<<8, 256, 0, stream>>>(z, AinvT, SZ);        // 64 waves
  gemm_sm_kernel<<<63, 256, 0, stream>>>(mu, AinvT, SM);      // 504 waves
  norms_kernel<<<141, 256, 0, stream>>>(SZ, SM, nz, nm);      // 1128 waves
  gemm_out_kernel<<<16, 256, 0, stream>>>(SZ, SM, nz, nm, prior, scal, out); // 128 waves
}